// MoBoAligner_90829968376033
// MI455X (gfx1250) — compile-verified
//
#include <hip/hip_runtime.h>
#include <math.h>

#define LOG_EPS (-1000.0f)

static constexpr int Bn = 4, In = 128, Jn = 1024, Tc = 512, Mc = 80, An = 128, Dn = 10;

typedef __attribute__((ext_vector_type(2))) float v2f;
typedef __attribute__((ext_vector_type(8))) float v8f;

__device__ __forceinline__ float logaddexpf_(float a, float b) {
  float m = fmaxf(a, b);
  float n = fminf(a, b);
  return m + log1pf(__expf(n - m));
}

// -----------------------------------------------------------------------------
// GEMM: out = X(rows x K) * W(cols x K)^T [+ bias], fp32 via V_WMMA_F32_16X16X4_F32.
// One wave32 per 16x16 output tile, looping K in steps of 4.
// A-frag layout (16x4 f32, ISA 7.12.2): lanes 0-15 hold M=lane, K={kb,kb+1};
// lanes 16-31 hold M=lane-16, K={kb+2,kb+3}. B-frag mirrored with N=lane&15.
// C/D: VGPR v holds row (v + 8*(lane>=16)), col = lane&15.
// transJ>0: output stored batched-transposed as out[b][col][j] with j-dim = transJ.
// -----------------------------------------------------------------------------
__global__ void wmma_gemm_nt(const float* __restrict__ X, const float* __restrict__ W,
                             const float* __restrict__ bias, float* __restrict__ out,
                             int rows, int K, int cols, int transJ) {
  int wave = (blockIdx.x * blockDim.x + threadIdx.x) >> 5;
  int lane = threadIdx.x & 31;
  int tilesPerRow = cols >> 4;
  int row0 = (wave / tilesPerRow) << 4;
  int col0 = (wave % tilesPerRow) << 4;
  if (row0 >= rows) return;
  int half = lane >> 4;   // 0: K pair {0,1}, 1: K pair {2,3}
  int lid  = lane & 15;

  v8f acc = {};
#if __has_builtin(__builtin_amdgcn_wmma_f32_16x16x4_f32)
  for (int kb = 0; kb < K; kb += 4) {
    int k0 = kb + (half << 1);
    float2 ax = *(const float2*)(X + (size_t)(row0 + lid) * K + k0);
    float2 bx = *(const float2*)(W + (size_t)(col0 + lid) * K + k0);
    v2f a; a.x = ax.x; a.y = ax.y;
    v2f b; b.x = bx.x; b.y = bx.y;
    // (neg_a, A, neg_b, B, c_mod, C, reuse_a, reuse_b)
    acc = __builtin_amdgcn_wmma_f32_16x16x4_f32(false, a, false, b, (short)0, acc,
                                                false, false);
  }
#else
  for (int k = 0; k < K; ++k) {
    float wv = W[(size_t)(col0 + lid) * K + k];
#pragma unroll
    for (int v = 0; v < 8; ++v)
      acc[v] += X[(size_t)(row0 + v + (half << 3)) * K + k] * wv;
  }
#endif

#pragma unroll
  for (int v = 0; v < 8; ++v) {
    int row = row0 + v + (half << 3);
    int col = col0 + lid;
    float val = acc[v] + (bias ? bias[col] : 0.0f);
    if (transJ) {
      int b = row / transJ;
      int j = row - b * transJ;
      out[((size_t)b * cols + col) * transJ + j] = val;
    } else {
      out[(size_t)row * cols + col] = val;
    }
  }
}

// -----------------------------------------------------------------------------
// e[b,i,j] = -softplus( -(vw . tanh(pt[b,i,:] + pm[b,j,:]) + vb) - noise[b,i,j] )
// pmT layout (B, A, J) so per-a reads are coalesced across j.
// -----------------------------------------------------------------------------
__global__ void energy_kernel(const float* __restrict__ pt, const float* __restrict__ pmT,
                              const float* __restrict__ vw, const float* __restrict__ vb,
                              const float* __restrict__ noise, float* __restrict__ e) {
  __shared__ float pts[An];
  __shared__ float vws[An];
  const int jblocks = Jn / 256;
  int bi = blockIdx.x / jblocks;           // b*I + i
  int jb = blockIdx.x % jblocks;
  int b  = bi / In;
  int tid = threadIdx.x;
  if (tid < An) {
    pts[tid] = pt[(size_t)bi * An + tid];
    vws[tid] = vw[tid];
  }
  __syncthreads();
  int j = jb * 256 + tid;
  const float* pmcol = pmT + (size_t)b * An * Jn + j;
  float s = vb[0];
#pragma unroll 4
  for (int a = 0; a < An; ++a)
    s += vws[a] * tanhf(pts[a] + pmcol[(size_t)a * Jn]);
  float t = -s - noise[(size_t)bi * Jn + j];
  float sp = fmaxf(t, 0.0f) + log1pf(__expf(-fabsf(t)));
  e[(size_t)bi * Jn + j] = -sp;
}

// -----------------------------------------------------------------------------
// lcp[b,i,d,k] = masked log P(dur=d+1 | boundary=k); geq = suffix-lse over d.
// One thread per (b,i,k); D=10 lives in registers.
// -----------------------------------------------------------------------------
__global__ void lcp_geq_kernel(const float* __restrict__ e, float* __restrict__ lcp,
                               float* __restrict__ geq) {
  int idx = blockIdx.x * blockDim.x + threadIdx.x;     // over B*I*J
  int b   = idx / (In * Jn);
  int rem = idx % (In * Jn);
  int i   = rem / Jn;
  int k   = rem % Jn;
  const float* erow = e + (size_t)(b * In + i) * Jn;
  int lim = Jn - In + i + 1;                           // k+d <= lim

  float ev[Dn];
  bool  va[Dn];
#pragma unroll
  for (int d = 0; d < Dn; ++d) {
    bool v = (k >= i) && (k + d <= lim) && (k + d < Jn);
    va[d] = v;
    ev[d] = v ? erow[k + d] : LOG_EPS;
  }
  float m = ev[0];
#pragma unroll
  for (int d = 1; d < Dn; ++d) m = fmaxf(m, ev[d]);
  float s = 0.0f;
#pragma unroll
  for (int d = 0; d < Dn; ++d) s += __expf(ev[d] - m);
  float l = m + __logf(s);

  float lc[Dn];
#pragma unroll
  for (int d = 0; d < Dn; ++d) lc[d] = va[d] ? (ev[d] - l) : LOG_EPS;

  size_t base = ((size_t)(b * In + i) * Dn) * Jn + k;
#pragma unroll
  for (int d = 0; d < Dn; ++d) lcp[base + (size_t)d * Jn] = lc[d];

  float acc = lc[Dn - 1];
  geq[base + (size_t)(Dn - 1) * Jn] = va[Dn - 1] ? acc : LOG_EPS;
#pragma unroll
  for (int d = Dn - 2; d >= 0; --d) {
    acc = logaddexpf_(acc, lc[d]);
    geq[base + (size_t)d * Jn] = va[d] ? acc : LOG_EPS;
  }
}

// -----------------------------------------------------------------------------
// Boundary DP: alpha[i,j] = lse_d alpha[i-1, j-d-1] + lcp[i-1, d, j-d-1].
// One workgroup (1024 thr = 32 waves) per batch; LDS double buffer; 127 steps.
// Stores alpha rows 0..I-1 (stride J+1) for the interval pass.
// -----------------------------------------------------------------------------
__global__ void dp_kernel(const float* __restrict__ lcp, float* __restrict__ alpha) {
  __shared__ float buf0[Jn + 1];
  __shared__ float buf1[Jn + 1];
  int b = blockIdx.x;
  int tid = threadIdx.x;
  float* prev = buf0;
  float* next = buf1;

  for (int j = tid; j <= Jn; j += blockDim.x) {
    float v = (j == 0) ? 0.0f : LOG_EPS;
    prev[j] = v;
    alpha[((size_t)b * In + 0) * (Jn + 1) + j] = v;
  }
  __syncthreads();

  for (int i = 1; i < In; ++i) {
    const float* lrow = lcp + ((size_t)(b * In + (i - 1)) * Dn) * Jn;
    for (int j = tid; j <= Jn; j += blockDim.x) {
      float nv = LOG_EPS;
      if (j >= i) {
        float vals[Dn];
#pragma unroll
        for (int d = 0; d < Dn; ++d) {
          int kk = j - d - 1;
          vals[d] = (kk >= 0) ? (prev[kk] + lrow[(size_t)d * Jn + kk]) : LOG_EPS;
        }
        float m = vals[0];
#pragma unroll
        for (int d = 1; d < Dn; ++d) m = fmaxf(m, vals[d]);
        float s = 0.0f;
#pragma unroll
        for (int d = 0; d < Dn; ++d) s += __expf(vals[d] - m);
        nv = m + __logf(s);
      }
      next[j] = nv;
      alpha[((size_t)b * In + i) * (Jn + 1) + j] = nv;
    }
    __syncthreads();
    float* t = prev; prev = next; next = t;
  }
}

// interval[b,r,jf] = lse_d alpha[b,r,jf-d] + geq[b,r,d,jf-d]   (r < I-1)
__global__ void interval_kernel(const float* __restrict__ alpha,
                                const float* __restrict__ geq,
                                float* __restrict__ out) {
  int idx = blockIdx.x * blockDim.x + threadIdx.x;   // over B*(I-1)*J
  int b   = idx / ((In - 1) * Jn);
  int rem = idx % ((In - 1) * Jn);
  int r   = rem / Jn;
  int jf  = rem % Jn;
  const float* arow = alpha + ((size_t)b * In + r) * (Jn + 1);
  const float* grow = geq + ((size_t)(b * In + r) * Dn) * Jn;
  float vals[Dn];
#pragma unroll
  for (int d = 0; d < Dn; ++d) {
    int k = jf - d;
    vals[d] = (k >= 0) ? (arow[k] + grow[(size_t)d * Jn + k]) : LOG_EPS;
  }
  float m = vals[0];
#pragma unroll
  for (int d = 1; d < Dn; ++d) m = fmaxf(m, vals[d]);
  float s = 0.0f;
#pragma unroll
  for (int d = 0; d < Dn; ++d) s += __expf(vals[d] - m);
  out[((size_t)b * In + r) * Jn + jf] = m + __logf(s);
}

// Last text token row: inclusive prefix cumlogsumexp of alpha[b, I-1, :J]
// via Hillis-Steele log-space scan in LDS (1024 threads per batch).
__global__ void lastrow_kernel(const float* __restrict__ alpha, float* __restrict__ out) {
  __shared__ float s[Jn];
  int b = blockIdx.x, tid = threadIdx.x;
  s[tid] = alpha[((size_t)b * In + (In - 1)) * (Jn + 1) + tid];
  __syncthreads();
  for (int off = 1; off < Jn; off <<= 1) {
    float v = s[tid];
    if (tid >= off) v = logaddexpf_(s[tid - off], v);
    __syncthreads();
    s[tid] = v;
    __syncthreads();
  }
  out[((size_t)b * In + (In - 1)) * Jn + tid] = s[tid];
}

extern "C" void kernel_launch(void* const* d_in, const int* in_sizes, int n_in,
                              void* d_out, int out_size, void* d_ws, size_t ws_size,
                              hipStream_t stream) {
  (void)in_sizes; (void)n_in; (void)out_size; (void)ws_size;
  const float* text  = (const float*)d_in[0];  // (B, I, Tc)
  const float* mel   = (const float*)d_in[1];  // (B, J, Mc)
  const float* Wm    = (const float*)d_in[2];  // (A, Mc)
  const float* bm    = (const float*)d_in[3];  // (A,)
  const float* Wt    = (const float*)d_in[4];  // (A, Tc)
  const float* vw    = (const float*)d_in[5];  // (A,)
  const float* vb    = (const float*)d_in[6];  // (1,)
  const float* noise = (const float*)d_in[7];  // (B, I, J)
  float* out = (float*)d_out;                  // (B, I, J)

  float* ws    = (float*)d_ws;
  float* pt    = ws;                                    // B*I*A
  float* pmT   = pt  + (size_t)Bn * In * An;            // B*A*J (transposed)
  float* e     = pmT + (size_t)Bn * An * Jn;            // B*I*J
  float* lcp   = e   + (size_t)Bn * In * Jn;            // B*I*D*J
  float* geq   = lcp + (size_t)Bn * In * Dn * Jn;       // B*I*D*J
  float* alpha = geq + (size_t)Bn * In * Dn * Jn;       // B*I*(J+1)

  // pt = text @ Wt^T : rows=512, K=512, cols=128 -> 256 tiles, 8 waves/block
  {
    int rows = Bn * In, K = Tc, cols = An;
    int tiles = (rows / 16) * (cols / 16);
    wmma_gemm_nt<<<tiles / 8, 256, 0, stream>>>(text, Wt, nullptr, pt, rows, K, cols, 0);
  }
  // pmT = (mel @ Wm^T + bm) stored (B, A, J) : rows=4096, K=80, cols=128
  {
    int rows = Bn * Jn, K = Mc, cols = An;
    int tiles = (rows / 16) * (cols / 16);
    wmma_gemm_nt<<<tiles / 8, 256, 0, stream>>>(mel, Wm, bm, pmT, rows, K, cols, Jn);
  }
  energy_kernel<<<Bn * In * (Jn / 256), 256, 0, stream>>>(pt, pmT, vw, vb, noise, e);
  lcp_geq_kernel<<<(Bn * In * Jn) / 256, 256, 0, stream>>>(e, lcp, geq);
  dp_kernel<<<Bn, 1024, 0, stream>>>(lcp, alpha);
  interval_kernel<<<(Bn * (In - 1) * Jn) / 256, 256, 0, stream>>>(alpha, geq, out);
  lastrow_kernel<<<Bn, Jn, 0, stream>>>(alpha, out);
}